// PseudoLabelGenerator2d_29703993819363
// MI455X (gfx1250) — compile-verified
//
#include <hip/hip_runtime.h>

// Problem constants from the reference
#define KJ   21          // joints
#define HW   4096        // 64*64 spatial
#define NPOS_TILE 16     // WMMA M/N tile of spatial positions

typedef __attribute__((ext_vector_type(2))) float v2f;
typedef __attribute__((ext_vector_type(8))) float v8f;

// ---------------------------------------------------------------------------
// Kernel 1: per-(b,k) argmax over 4096 elements -> heatmap slice offset
// ---------------------------------------------------------------------------
__global__ __launch_bounds__(256) void argmax_rows_kernel(
    const float* __restrict__ y, int* __restrict__ slice_off)
{
    __shared__ float sv[256];
    __shared__ int   si[256];

    const int row = blockIdx.x;                 // row = b*21 + k
    const float* src = y + (size_t)row * HW;
    const int t = threadIdx.x;

    float best = -__builtin_inff();
    int   bidx = 0;
    // strided coalesced scan; strict '>' keeps first occurrence per thread
    for (int i = t; i < HW; i += 256) {
        float v = src[i];
        if (v > best) { best = v; bidx = i; }
    }
    sv[t] = best; si[t] = bidx;
    __syncthreads();

    for (int s = 128; s > 0; s >>= 1) {
        if (t < s) {
            float ov = sv[t + s]; int oi = si[t + s];
            // tie-break: lowest linear index (matches jnp.argmax)
            if (ov > sv[t] || (ov == sv[t] && oi < si[t])) { sv[t] = ov; si[t] = oi; }
        }
        __syncthreads();
    }
    if (t == 0) {
        int idx = (sv[0] > 0.0f) ? si[0] : 0;   // pred_mask: invalid -> (0,0)
        int px = idx & 63;                      // idx % W
        int py = idx >> 6;                      // idx / W
        slice_off[row] = ((px << 6) + py) << 12; // (px*64 + py) * 4096 elems
    }
}

// ---------------------------------------------------------------------------
// Kernel 2: gather gt slices, emit ground_truth, and compute
//   gf[b,j,p] = clip( sum_k gt[b,k,p] * FM[k,j], 0, 1 )
// as D = A x B with A = FM^T (M=j, K=k), B = gt (K=k, N=position),
// using V_WMMA_F32_16X16X4_F32 (full fp32, matches reference numerics).
// One wave handles one (b, 16-position tile). 6 K-steps x 2 j-tiles = 12 WMMA.
// ---------------------------------------------------------------------------
__global__ __launch_bounds__(256) void pseudo_label_kernel(
    const float* __restrict__ heat,     // (64,64,64,64) table, L2-resident
    const float* __restrict__ fm,       // (21,21) false_matrix
    const int*   __restrict__ slice_off,// (B*21) element offsets into heat
    float* __restrict__ out_gt,         // (B,21,64,64)
    float* __restrict__ out_gf)         // (B,21,64,64)
{
    // FM staged in LDS, zero-padded to 24 x 32 so K->24, N->32 padding is
    // branch-free (rows k>=21 and cols j>=21 contribute exactly 0).
    __shared__ float sFM[24 * 32];

    const int t = threadIdx.x;
    for (int i = t; i < 24 * 32; i += 256) sFM[i] = 0.0f;
    __syncthreads();
    for (int i = t; i < KJ * KJ; i += 256) {
        int k = i / KJ, j = i % KJ;
        sFM[k * 32 + j] = fm[i];
    }
    __syncthreads();

    const int wave  = (blockIdx.x * 256 + t) >> 5;   // global wave id
    const int lane  = t & 31;
    const int b     = wave >> 8;                     // 256 position-tiles per b
    const int ptile = wave & 255;
    const int pbase = ptile << 4;

    const int half = lane >> 4;      // 0: K0/K1 half, 1: K2/K3 half
    const int q    = lane & 15;      // A: row m (=j in tile); B/D: column n (=pos)
    const int p    = pbase + q;

    const int*   offb = slice_off + b * KJ;
    float* __restrict__ gt_b = out_gt + (size_t)b * KJ * HW;
    float* __restrict__ gf_b = out_gf + (size_t)b * KJ * HW;

    v8f c0 = {};   // j-tile 0: j = 0..15
    v8f c1 = {};   // j-tile 1: j = 16..31 (only 16..20 stored)

    for (int k0 = 0; k0 < 24; k0 += 4) {
        const int kb = k0 + half * 2;            // this lane's k pair: kb, kb+1

        // ---- B tile (gt, 4x16): v0 = row kb, v1 = row kb+1 at column p ----
        float b0 = 0.0f, b1 = 0.0f;
        if (kb < KJ)     b0 = heat[(size_t)offb[kb]     + p];
        if (kb + 1 < KJ) b1 = heat[(size_t)offb[kb + 1] + p];
        // each (k,p) is gathered exactly once across the wave -> also emit gt
        if (kb < KJ)     gt_b[kb * HW + p]       = b0;
        if (kb + 1 < KJ) gt_b[(kb + 1) * HW + p] = b1;
        v2f Bm; Bm.x = b0; Bm.y = b1;

        // ---- A tiles (FM^T, 16x4): A[m][kk] = FM[k][ jt*16 + m ] ----
        v2f A0, A1;
        A0.x = sFM[kb * 32 + q];            A0.y = sFM[(kb + 1) * 32 + q];
        A1.x = sFM[kb * 32 + 16 + q];       A1.y = sFM[(kb + 1) * 32 + 16 + q];

        // (neg_a, A, neg_b, B, c_mod, C, reuse_a, reuse_b)
        c0 = __builtin_amdgcn_wmma_f32_16x16x4_f32(
                 false, A0, false, Bm, (short)0, c0, false, false);
        c1 = __builtin_amdgcn_wmma_f32_16x16x4_f32(
                 false, A1, false, Bm, (short)0, c1, false, false);
    }

    // ---- store D: VGPR r holds row (half*8 + r), column = q (position) ----
    const int mb = half * 8;
    #pragma unroll
    for (int r = 0; r < 8; ++r) {
        // j-tile 0: j = mb + r  (always < 16 < 21)
        float v0 = c0[r];
        v0 = fminf(fmaxf(v0, 0.0f), 1.0f);
        gf_b[(mb + r) * HW + pbase + q] = v0;

        // j-tile 1: j = 16 + mb + r (valid only for 16..20)
        int j1 = 16 + mb + r;
        if (j1 < KJ) {
            float v1 = c1[r];
            v1 = fminf(fmaxf(v1, 0.0f), 1.0f);
            gf_b[j1 * HW + pbase + q] = v1;
        }
    }
}

// ---------------------------------------------------------------------------
extern "C" void kernel_launch(void* const* d_in, const int* in_sizes, int n_in,
                              void* d_out, int out_size, void* d_ws, size_t ws_size,
                              hipStream_t stream)
{
    const float* y    = (const float*)d_in[0];   // (B,21,64,64)
    const float* heat = (const float*)d_in[1];   // (64,64,64,64)
    const float* fm   = (const float*)d_in[2];   // (21,21)

    const int B = in_sizes[0] / (KJ * HW);       // 256

    float* out_gt = (float*)d_out;
    float* out_gf = out_gt + (size_t)B * KJ * HW;
    int*   off    = (int*)d_ws;                  // B*21 ints of scratch

    // Kernel 1: one block per (b,k) row
    argmax_rows_kernel<<<B * KJ, 256, 0, stream>>>(y, off);

    // Kernel 2: one wave per (b, 16-position tile); 8 waves per block
    const int total_waves = B * (HW / NPOS_TILE);        // B * 256
    pseudo_label_kernel<<<total_waves / 8, 256, 0, stream>>>(
        heat, fm, off, out_gt, out_gf);
}